// DeformableConvAttention_41824391529103
// MI455X (gfx1250) — compile-verified
//
#include <hip/hip_runtime.h>
#include <hip/hip_bf16.h>
#include <math.h>

typedef float v2f __attribute__((ext_vector_type(2)));
typedef float v8f __attribute__((ext_vector_type(8)));
typedef unsigned int v4u __attribute__((ext_vector_type(4)));
typedef unsigned int v8u __attribute__((ext_vector_type(8)));

#define N_HEADS 8
#define DH      64
#define G_      4
#define KPTS    9
#define C_      512
#define CG_     128
#define H_      64
#define W_      64
#define HW      4096
#define SCALE_  0.125f   // DH^-0.5

#define KB_     32       // K-chunk staged per TDM round
#define BN_     256      // block N tile
#define BM_     32       // block M tile

// ---------------------------------------------------------------------------
// Issue one TDM 2D tile load (rows x cols of f32, row stride = ldN elements)
// from global memory into LDS. 2-SGPR-group form (tensor <= 2D), per CDNA5 ISA
// ch.8 descriptor layout. Issued once per wave; tracked with TENSORcnt.
// ---------------------------------------------------------------------------
__device__ __forceinline__ void tdm_load_tile_f32(const float* gsrc,
                                                  unsigned lds_byte_addr,
                                                  int cols, int rows, int ldN)
{
    unsigned long long ga = (unsigned long long)(uintptr_t)gsrc;
    v4u g0;
    g0.x = 1u;                                   // count=1, no gather
    g0.y = lds_byte_addr;                        // D#.lds_addr
    g0.z = (unsigned)(ga & 0xffffffffu);         // D#.global_addr[31:0]
    g0.w = (unsigned)((ga >> 32) & 0x01ffffffu)  // D#.global_addr[56:32]
         | 0x80000000u;                          // type=2 ("image")
    v8u g1;
    const unsigned tensor_d0 = 1u << 20;         // large: no OOB clamping
    const unsigned tensor_d1 = 1u << 20;
    g1.s0 = 0x20000u;                            // workgroup_mask=0, data_size=2 (4B)
    g1.s1 = (tensor_d0 & 0xffffu) << 16;         // tensor_dim0[15:0]
    g1.s2 = (tensor_d0 >> 16) | ((tensor_d1 & 0xffffu) << 16);
    g1.s3 = (tensor_d1 >> 16) | ((unsigned)cols << 16);   // tile_dim0
    g1.s4 = (unsigned)rows;                      // tile_dim1 (tile_dim2=0)
    g1.s5 = (unsigned)ldN;                       // tensor_dim0_stride[31:0]
    g1.s6 = 0u;                                  // stride hi=0, dim1_stride lo=0
    g1.s7 = 0u;
    asm volatile("tensor_load_to_lds %0, %1" :: "s"(g0), "s"(g1) : "memory");
}

// ---------------------------------------------------------------------------
// f32 WMMA GEMM: C[MxN] = A[MxK] * B[KxN] + bias[M], row-major; optional
// transposed store C[n*M+m].  Block = 8 waves covering 32(M) x 256(N).
// B is staged through LDS by the Tensor Data Mover in 32x256 chunks,
// double-buffered: each wave DMAs its own 4x256 slice and waits TENSORcnt.
// Wave tile 16x64 -> 4 f32 WMMA accumulators, A-fragment reuse.
// Requires M%32==0, N%256==0, K%32==0.
// f32 fragment layouts per CDNA5 ISA 7.12.2:
//   A 16x4 : lanes 0-15 row M=lane (v0:K=0, v1:K=1); lanes 16-31 (v0:K=2, v1:K=3)
//   B 4x16 : lanes 0-15 col N=lane (v0:K=0, v1:K=1); lanes 16-31 (v0:K=2, v1:K=3)
//   C/D    : vgpr r: lanes 0-15 -> (M=r, N=lane); lanes 16-31 -> (M=r+8, N=lane-16)
// ---------------------------------------------------------------------------
__global__ void gemm_wmma_f32(const float* __restrict__ A,
                              const float* __restrict__ B,
                              const float* __restrict__ bias,
                              float* __restrict__ Cmat,
                              int M, int N, int K, int transpose_out)
{
    __shared__ float Bs[2][KB_ * BN_];           // 2 x 32KB double buffer

    const int wave = threadIdx.x >> 5;
    const int lane = threadIdx.x & 31;
    const int nBlocks = N >> 8;
    const int bm = blockIdx.x / nBlocks;
    const int bn = blockIdx.x - bm * nBlocks;
    const int m0 = bm * BM_ + (wave >> 2) * 16;
    const int n0l = (wave & 3) * 64;             // N offset inside block tile
    const int n0 = bn * BN_ + n0l;

    const int half  = lane >> 4;                 // 0: K+{0,1}, 1: K+{2,3}
    const int l16   = lane & 15;
    const int arow  = m0 + l16;
    const int akoff = half * 2;

    const unsigned ldsBase = (unsigned)(uintptr_t)(void*)&Bs[0][0];
    const unsigned sliceOff = (unsigned)(wave * 4 * BN_ * 4);   // wave's 4 rows
    const int nChunks = K / KB_;

    // prologue: every wave DMAs its 4x256 slice of chunk 0 into buffer 0
    tdm_load_tile_f32(B + (size_t)(wave * 4) * N + bn * BN_,
                      ldsBase + sliceOff, BN_, 4, N);

    v8f acc0 = {0.f,0.f,0.f,0.f,0.f,0.f,0.f,0.f};
    v8f acc1 = acc0, acc2 = acc0, acc3 = acc0;

    for (int cb = 0; cb < nChunks; ++cb) {
        const int buf = cb & 1;
        if (cb + 1 < nChunks) {
            tdm_load_tile_f32(B + (size_t)((cb + 1) * KB_ + wave * 4) * N + bn * BN_,
                              ldsBase + (unsigned)((buf ^ 1) * KB_ * BN_ * 4) + sliceOff,
                              BN_, 4, N);
            __builtin_amdgcn_s_wait_tensorcnt(1);   // own slice of chunk cb landed
        } else {
            __builtin_amdgcn_s_wait_tensorcnt(0);
        }
        __syncthreads();                            // all slices of chunk cb visible

        const float* __restrict__ Bl = &Bs[buf][0];
        #pragma unroll
        for (int ks = 0; ks < KB_ / 4; ++ks) {
            const int kglob = cb * KB_ + ks * 4 + akoff;
            v2f a;
            a.x = A[(size_t)arow * K + kglob];
            a.y = A[(size_t)arow * K + kglob + 1];

            const int r0 = (ks * 4 + akoff) * BN_;
            const int c0 = n0l + l16;
            v2f b0, b1, b2, b3;
            b0.x = Bl[r0 + c0];            b0.y = Bl[r0 + BN_ + c0];
            b1.x = Bl[r0 + c0 + 16];       b1.y = Bl[r0 + BN_ + c0 + 16];
            b2.x = Bl[r0 + c0 + 32];       b2.y = Bl[r0 + BN_ + c0 + 32];
            b3.x = Bl[r0 + c0 + 48];       b3.y = Bl[r0 + BN_ + c0 + 48];

            acc0 = __builtin_amdgcn_wmma_f32_16x16x4_f32(false, a, false, b0, (short)0, acc0, false, false);
            acc1 = __builtin_amdgcn_wmma_f32_16x16x4_f32(false, a, false, b1, (short)0, acc1, false, false);
            acc2 = __builtin_amdgcn_wmma_f32_16x16x4_f32(false, a, false, b2, (short)0, acc2, false, false);
            acc3 = __builtin_amdgcn_wmma_f32_16x16x4_f32(false, a, false, b3, (short)0, acc3, false, false);
        }
        __syncthreads();                            // protect buf before re-DMA
    }

    #pragma unroll
    for (int r = 0; r < 8; ++r) {
        const int m  = m0 + r + half * 8;
        const float bv = bias[m];
        const float vals[4] = {acc0[r], acc1[r], acc2[r], acc3[r]};
        #pragma unroll
        for (int t = 0; t < 4; ++t) {
            const int col = n0 + l16 + t * 16;
            const float v = vals[t] + bv;
            if (!transpose_out) Cmat[(size_t)m * N + col] = v;
            else                Cmat[(size_t)col * M + m] = v;
        }
    }
}

// ---------------------------------------------------------------------------
// Depthwise 3x3 (zero-pad SAME) + LayerNorm over CG channels + exact GELU.
// Input/output layout (G*CG, H, W). One block of 128 threads per (g, pixel).
// ---------------------------------------------------------------------------
__global__ void dwconv_ln_gelu(const float* __restrict__ in,
                               const float* __restrict__ w,     // (CG,3,3)
                               const float* __restrict__ wb,    // (CG)
                               const float* __restrict__ gamma,
                               const float* __restrict__ beta,
                               float* __restrict__ out)
{
    __shared__ float s_sum[CG_];
    __shared__ float s_sq[CG_];
    const int cg  = threadIdx.x;          // 0..127
    const int blk = blockIdx.x;           // g*HW + p
    const int g   = blk >> 12;            // / 4096
    const int p   = blk & (HW - 1);
    const int h   = p >> 6;
    const int wc  = p & 63;

    const float* plane = in + (size_t)(g * CG_ + cg) * HW;
    float acc = wb[cg];
    #pragma unroll
    for (int ky = 0; ky < 3; ++ky) {
        const int y = h + ky - 1;
        if (y < 0 || y >= H_) continue;
        #pragma unroll
        for (int kx = 0; kx < 3; ++kx) {
            const int xcol = wc + kx - 1;
            if (xcol < 0 || xcol >= W_) continue;
            acc += w[cg * 9 + ky * 3 + kx] * plane[y * W_ + xcol];
        }
    }
    s_sum[cg] = acc;
    s_sq[cg]  = acc * acc;
    __syncthreads();
    for (int s = 64; s > 0; s >>= 1) {
        if (cg < s) { s_sum[cg] += s_sum[cg + s]; s_sq[cg] += s_sq[cg + s]; }
        __syncthreads();
    }
    const float mean = s_sum[0] * (1.0f / CG_);
    const float var  = s_sq[0]  * (1.0f / CG_) - mean * mean;
    const float xn   = (acc - mean) * rsqrtf(var + 1e-5f) * gamma[cg] + beta[cg];
    const float ge   = 0.5f * xn * (1.0f + erff(xn * 0.70710678f));
    out[(size_t)(g * CG_ + cg) * HW + p] = ge;
}

// ---------------------------------------------------------------------------
// Offset head at selected pixels only: for each (g, j, k) compute the 2 dot
// products (length 128) against ow3, tanh-scale, add reference grid, emit
// unnormalized sample coordinates (gx, gy).
// ---------------------------------------------------------------------------
__global__ void offset_grid(const float* __restrict__ t2,    // (G*CG, HW)
                            const float* __restrict__ ow3,   // (2K, CG)
                            const int*   __restrict__ idx,   // (n,2)
                            float* __restrict__ grid_sel,    // (G*K, n, 2)
                            int n)
{
    const int tid = blockIdx.x * blockDim.x + threadIdx.x;
    const int total = G_ * KPTS * n;
    if (tid >= total) return;
    const int k = tid % KPTS;
    const int j = (tid / KPTS) % n;
    const int g = tid / (KPTS * n);
    const int iy = idx[2 * j], ix = idx[2 * j + 1];
    const int p  = iy * W_ + ix;

    const float* tp = t2 + (size_t)g * CG_ * HW + p;
    const float* wy = ow3 + (2 * k) * CG_;
    const float* wx = ow3 + (2 * k + 1) * CG_;
    float oy = 0.f, ox = 0.f;
    for (int c = 0; c < CG_; ++c) {
        const float tv = tp[(size_t)c * HW];
        oy += wy[c] * tv;
        ox += wx[c] * tv;
    }
    const int dy = k / 3, dx = k % 3;
    const int ry = min(max(iy + dy - 1, 0), H_ - 1);
    const int rx = min(max(ix + dx - 1, 0), W_ - 1);
    const float refy = ((float)ry + 0.5f) * (2.0f / H_) - 1.0f;
    const float refx = ((float)rx + 0.5f) * (2.0f / W_) - 1.0f;
    const float py = tanhf(oy) * (2.0f / H_) + refy;   // rng * FACTOR = (1/H)*2
    const float px = tanhf(ox) * (2.0f / W_) + refx;
    const float gx = (px + 1.0f) * 0.5f * (W_ - 1);
    const float gy = (py + 1.0f) * 0.5f * (H_ - 1);
    const size_t o = ((size_t)(g * KPTS + k) * n + j) * 2;
    grid_sel[o]     = gx;
    grid_sel[o + 1] = gy;
}

// ---------------------------------------------------------------------------
// Fused bilinear gather producing BOTH outputs:
//   xr_out[j][gk][c] for all 512 channels (coalesced over c)
//   xs_sel[c][k][j]  when channel c belongs to group g (subset of xr values)
// ---------------------------------------------------------------------------
__global__ void sample_kernel(const float* __restrict__ x,        // (C, HW)
                              const float* __restrict__ grid_sel, // (G*K, n, 2)
                              float* __restrict__ xr_out,         // (n, 36, C)
                              float* __restrict__ xs_sel,         // (C, K, n)
                              int n)
{
    const size_t tid = (size_t)blockIdx.x * blockDim.x + threadIdx.x;
    const size_t total = (size_t)G_ * KPTS * n * C_;
    if (tid >= total) return;
    const int c  = (int)(tid % C_);
    const int j  = (int)((tid / C_) % n);
    const int gk = (int)(tid / ((size_t)C_ * n));

    const size_t go = ((size_t)gk * n + j) * 2;
    const float gx = grid_sel[go];
    const float gy = grid_sel[go + 1];
    const float x0f = floorf(gx), y0f = floorf(gy);
    const int   x0 = (int)x0f,    y0 = (int)y0f;
    const float wx1 = gx - x0f, wx0 = 1.0f - wx1;
    const float wy1 = gy - y0f, wy0 = 1.0f - wy1;

    const float* plane = x + (size_t)c * HW;
    float s = 0.f;
    if (y0 >= 0     && y0 < H_     && x0 >= 0     && x0 < W_    ) s += plane[y0 * W_ + x0]             * wy0 * wx0;
    if (y0 >= 0     && y0 < H_     && x0 + 1 >= 0 && x0 + 1 < W_) s += plane[y0 * W_ + x0 + 1]         * wy0 * wx1;
    if (y0 + 1 >= 0 && y0 + 1 < H_ && x0 >= 0     && x0 < W_    ) s += plane[(y0 + 1) * W_ + x0]       * wy1 * wx0;
    if (y0 + 1 >= 0 && y0 + 1 < H_ && x0 + 1 >= 0 && x0 + 1 < W_) s += plane[(y0 + 1) * W_ + x0 + 1]   * wy1 * wx1;

    xr_out[((size_t)j * (G_ * KPTS) + gk) * C_ + c] = s;

    const int g = gk / KPTS, k = gk - g * KPTS;
    if ((c >> 7) == g) {
        xs_sel[((size_t)c * KPTS + k) * n + j] = s;
    }
}

// ---------------------------------------------------------------------------
// Attention: per (head, pixel): 9 dots of length 64, softmax over 9,
// weighted sum of v. attn written straight to d_out, out_buf to workspace.
// ---------------------------------------------------------------------------
__global__ void attention_kernel(const float* __restrict__ q_full, // (C, HW)
                                 const float* __restrict__ kmat,   // (C, K, n)
                                 const float* __restrict__ vmat,   // (C, K, n)
                                 const int*   __restrict__ idx,
                                 float* __restrict__ attn_out,     // (8, n, 9)
                                 float* __restrict__ out_buf,      // (C, n)
                                 int n)
{
    const int tid = blockIdx.x * blockDim.x + threadIdx.x;
    if (tid >= N_HEADS * n) return;
    const int j = tid % n;
    const int h = tid / n;
    const int p = idx[2 * j] * W_ + idx[2 * j + 1];

    float att[KPTS];
    #pragma unroll
    for (int kk = 0; kk < KPTS; ++kk) att[kk] = 0.f;

    for (int d = 0; d < DH; ++d) {
        const float qv = q_full[(size_t)(h * DH + d) * HW + p];
        const float* kr = kmat + (size_t)(h * DH + d) * KPTS * n + j;
        #pragma unroll
        for (int kk = 0; kk < KPTS; ++kk)
            att[kk] += qv * kr[(size_t)kk * n];
    }
    float mx = -INFINITY;
    #pragma unroll
    for (int kk = 0; kk < KPTS; ++kk) { att[kk] *= SCALE_; mx = fmaxf(mx, att[kk]); }
    float ssum = 0.f;
    #pragma unroll
    for (int kk = 0; kk < KPTS; ++kk) { att[kk] = expf(att[kk] - mx); ssum += att[kk]; }
    const float inv = 1.0f / ssum;
    #pragma unroll
    for (int kk = 0; kk < KPTS; ++kk) {
        att[kk] *= inv;
        attn_out[((size_t)h * n + j) * KPTS + kk] = att[kk];
    }
    for (int d = 0; d < DH; ++d) {
        const float* vr = vmat + (size_t)(h * DH + d) * KPTS * n + j;
        float acc = 0.f;
        #pragma unroll
        for (int kk = 0; kk < KPTS; ++kk) acc += att[kk] * vr[(size_t)kk * n];
        out_buf[(size_t)(h * DH + d) * n + j] = acc;
    }
}

__global__ void sigmoid_kernel(const float* __restrict__ y,   // (n, C)
                               const int*   __restrict__ labels,
                               float* __restrict__ sig, int n)
{
    const int j = blockIdx.x * blockDim.x + threadIdx.x;
    if (j >= n) return;
    const float v = y[(size_t)j * C_ + labels[j]];
    sig[j] = 1.0f / (1.0f + expf(-v));
}

// ---------------------------------------------------------------------------
extern "C" void kernel_launch(void* const* d_in, const int* in_sizes, int n_in,
                              void* d_out, int out_size, void* d_ws, size_t ws_size,
                              hipStream_t stream) {
    const float* x      = (const float*)d_in[0];
    const int*   idx    = (const int*)  d_in[1];
    const int*   labels = (const int*)  d_in[2];
    const float* wq  = (const float*)d_in[3];
    const float* bq  = (const float*)d_in[4];
    const float* wk  = (const float*)d_in[5];
    const float* bk  = (const float*)d_in[6];
    const float* wv  = (const float*)d_in[7];
    const float* bv  = (const float*)d_in[8];
    const float* wo  = (const float*)d_in[9];
    const float* bo  = (const float*)d_in[10];
    const float* ow1 = (const float*)d_in[11];
    const float* ob1 = (const float*)d_in[12];
    const float* g1  = (const float*)d_in[13];
    const float* b1  = (const float*)d_in[14];
    const float* ow2 = (const float*)d_in[15];
    const float* ob2 = (const float*)d_in[16];
    const float* g2  = (const float*)d_in[17];
    const float* b2  = (const float*)d_in[18];
    const float* ow3 = (const float*)d_in[19];

    const int n = in_sizes[1] / 2;          // 2048

    // ---- workspace layout (floats) ----
    float* ws = (float*)d_ws;
    float* q_full   = ws;                                   // C*HW
    float* t1       = q_full + (size_t)C_ * HW;             // C*HW
    float* t2       = t1     + (size_t)C_ * HW;             // C*HW
    float* grid_sel = t2     + (size_t)C_ * HW;             // G*K*n*2
    float* xs_sel   = grid_sel + (size_t)G_ * KPTS * n * 2; // C*K*n
    float* kmat     = xs_sel + (size_t)C_ * KPTS * n;       // C*K*n
    float* vmat     = kmat   + (size_t)C_ * KPTS * n;       // C*K*n
    float* out_buf  = vmat   + (size_t)C_ * KPTS * n;       // C*n

    // ---- output layout (floats) ----
    float* y_out    = (float*)d_out;                        // (n, C)
    float* attn_out = y_out + (size_t)n * C_;               // (8, n, 9)
    float* xr_out   = attn_out + (size_t)N_HEADS * n * KPTS;// (n, 36, C)
    float* sig_out  = xr_out + (size_t)n * G_ * KPTS * C_;  // (n)

    // 1) q = wq @ x + bq                    M=512 N=4096 K=512
    gemm_wmma_f32<<<(C_ / BM_) * (HW / BN_), 256, 0, stream>>>(
        wq, x, bq, q_full, C_, HW, C_, 0);

    // 2,3) depthwise+LN+GELU twice
    dwconv_ln_gelu<<<G_ * HW, CG_, 0, stream>>>(q_full, ow1, ob1, g1, b1, t1);
    dwconv_ln_gelu<<<G_ * HW, CG_, 0, stream>>>(t1,     ow2, ob2, g2, b2, t2);

    // 4) offsets -> sample coordinates at selected pixels
    {
        const int total = G_ * KPTS * n;
        offset_grid<<<(total + 255) / 256, 256, 0, stream>>>(t2, ow3, idx, grid_sel, n);
    }

    // 5) fused bilinear gather -> xr_out (d_out) and xs_sel (ws)
    {
        const size_t total = (size_t)G_ * KPTS * n * C_;
        sample_kernel<<<(unsigned)((total + 255) / 256), 256, 0, stream>>>(
            x, grid_sel, xr_out, xs_sel, n);
    }

    // 6,7) k = wk @ xs_sel + bk ; v = wv @ xs_sel + bv     M=512 N=9n K=512
    {
        const int N = KPTS * n;  // 18432
        const int grid = (C_ / BM_) * (N / BN_);
        gemm_wmma_f32<<<grid, 256, 0, stream>>>(wk, xs_sel, bk, kmat, C_, N, C_, 0);
        gemm_wmma_f32<<<grid, 256, 0, stream>>>(wv, xs_sel, bv, vmat, C_, N, C_, 0);
    }

    // 8) attention + weighted value sum
    attention_kernel<<<(N_HEADS * n + 255) / 256, 256, 0, stream>>>(
        q_full, kmat, vmat, idx, attn_out, out_buf, n);

    // 9) y = (wo @ out + bo)^T  stored directly as (n, C)
    gemm_wmma_f32<<<(C_ / BM_) * (n / BN_), 256, 0, stream>>>(
        wo, out_buf, bo, y_out, C_, n, C_, 1);

    // 10) sig = sigmoid(y[j, labels[j]])
    sigmoid_kernel<<<(n + 255) / 256, 256, 0, stream>>>(y_out, labels, sig_out, n);
}